// ElmanNetLayer_71219147702763
// MI455X (gfx1250) — compile-verified
//
#include <hip/hip_runtime.h>
#include <math.h>

// CDNA5 (gfx1250) Elman RNN layer.
//   Phase 1: x2h = x @ W^T + b           (massively parallel f32-WMMA GEMM)
//   Phase 2: h_t = tanh(x2h_t + h_{t-1} @ U^T)   (persistent kernel, grid barrier)
//
// V_WMMA_F32_16X16X4_F32 fragments (ISA 7.12.2, wave32):
//   A 16x4 : lane<16 -> M=lane, {v0,v1}={A[M][k+0],A[M][k+1]}
//            lane>=16-> M=lane-16, {v0,v1}={A[M][k+2],A[M][k+3]}
//   B 4x16 : lane<16 -> N=lane, {v0,v1}={B[k+0][N],B[k+1][N]}
//            lane>=16-> N=lane-16, {v0,v1}={B[k+2][N],B[k+3][N]}
//   C/D 16x16: vgpr r, lane<16 -> (M=r,   N=lane), lane>=16 -> (M=r+8, N=lane-16)

typedef __attribute__((ext_vector_type(2))) float v2f;
typedef __attribute__((ext_vector_type(8))) float v8f;

#define BDIM 64
#define SDIM 1024
#define KDIM 1024   // IN
#define ODIM 1024   // OUT
#define NBLK_SCAN 8 // persistent-scan blocks (8 blk x 8 waves = 64 n-tiles)

__device__ __forceinline__ v8f wmma_f32(v2f a, v2f b, v8f c) {
  return __builtin_amdgcn_wmma_f32_16x16x4_f32(false, a, false, b, (short)0, c,
                                               false, false);
}

// ---------------------------------------------------------------------------
// Phase 1: C[m][n] = sum_k X[m][k]*W[n][k] + bias[n],  M = B*S = 65536.
// grid = (8, M/64), block = 256 (8 waves). Wave computes a 64x16 tile of C.
// ---------------------------------------------------------------------------
__global__ __launch_bounds__(256) void x2h_kernel(
    const float* __restrict__ X, const float* __restrict__ W,
    const float* __restrict__ bias, float* __restrict__ C) {
  const int lane  = threadIdx.x & 31;
  const int wave  = threadIdx.x >> 5;
  const int khalf = lane >> 4;   // 0 -> k+0/1, 1 -> k+2/3
  const int lidx  = lane & 15;

  const int n0 = (blockIdx.x * 8 + wave) * 16;
  const int m0 = blockIdx.y * 64;

  // B = W^T  =>  B[k][n] = W[n][k]: contiguous along k per lane.
  const float* Wrow = W + (size_t)(n0 + lidx) * KDIM + 2 * khalf;
  const float* Xrow = X + (size_t)(m0 + lidx) * KDIM + 2 * khalf;

  v8f acc[4] = {v8f{}, v8f{}, v8f{}, v8f{}};

  for (int k = 0; k < KDIM; k += 4) {
    v2f bfrag = *(const v2f*)(Wrow + k);
    v2f a0 = *(const v2f*)(Xrow + k);
    v2f a1 = *(const v2f*)(Xrow + (size_t)16 * KDIM + k);
    v2f a2 = *(const v2f*)(Xrow + (size_t)32 * KDIM + k);
    v2f a3 = *(const v2f*)(Xrow + (size_t)48 * KDIM + k);
    acc[0] = wmma_f32(a0, bfrag, acc[0]);
    acc[1] = wmma_f32(a1, bfrag, acc[1]);
    acc[2] = wmma_f32(a2, bfrag, acc[2]);
    acc[3] = wmma_f32(a3, bfrag, acc[3]);
  }

  const float bv = bias[n0 + lidx];
#pragma unroll
  for (int mi = 0; mi < 4; ++mi) {
#pragma unroll
    for (int r = 0; r < 8; ++r) {
      const int row = m0 + mi * 16 + r + 8 * khalf;
      C[(size_t)row * ODIM + n0 + lidx] = acc[mi][r] + bv;
    }
  }
}

// ---------------------------------------------------------------------------
// Phase 2: persistent scan. out[:,t,:] holds x2h_t on entry, h_t on exit.
// grid = 8 blocks x 256 threads; wave w of block b owns n-tile (b*8+w).
// barrier[0] = arrive counter, barrier[1] = generation (zeroed per launch).
// ---------------------------------------------------------------------------
__global__ __launch_bounds__(256) void rnn_scan_kernel(
    const float* __restrict__ h0, const float* __restrict__ U,
    float* __restrict__ out, unsigned* __restrict__ barrier) {
  const int lane  = threadIdx.x & 31;
  const int wave  = threadIdx.x >> 5;
  const int khalf = lane >> 4;
  const int lidx  = lane & 15;
  const int n0    = (blockIdx.x * 8 + wave) * 16;

  unsigned* arrive = barrier;
  unsigned* gen    = barrier + 1;
  const unsigned nblk = gridDim.x;

  const float* Urow = U + (size_t)(n0 + lidx) * KDIM + 2 * khalf;

  for (int t = 0; t < SDIM; ++t) {
    const float* hprev;
    size_t hstride;
    if (t == 0) { hprev = h0;                          hstride = ODIM; }
    else        { hprev = out + (size_t)(t - 1) * ODIM; hstride = (size_t)SDIM * ODIM; }

    const float* Arow = hprev + (size_t)lidx * hstride + 2 * khalf;

    v8f acc[4] = {v8f{}, v8f{}, v8f{}, v8f{}};
    for (int k = 0; k < KDIM; k += 4) {
      v2f bfrag = *(const v2f*)(Urow + k);
      v2f a0 = *(const v2f*)(Arow + k);
      v2f a1 = *(const v2f*)(Arow + 16 * hstride + k);
      v2f a2 = *(const v2f*)(Arow + 32 * hstride + k);
      v2f a3 = *(const v2f*)(Arow + 48 * hstride + k);
      acc[0] = wmma_f32(a0, bfrag, acc[0]);
      acc[1] = wmma_f32(a1, bfrag, acc[1]);
      acc[2] = wmma_f32(a2, bfrag, acc[2]);
      acc[3] = wmma_f32(a3, bfrag, acc[3]);
    }

    // h_t = tanh(acc + x2h_t); read-modify-write out[:, t, :] in place.
#pragma unroll
    for (int mi = 0; mi < 4; ++mi) {
#pragma unroll
      for (int r = 0; r < 8; ++r) {
        const int row  = mi * 16 + r + 8 * khalf;   // batch index
        const size_t i = ((size_t)row * SDIM + t) * ODIM + n0 + lidx;
        out[i] = tanhf(acc[mi][r] + out[i]);
      }
    }

    // ---- device-wide barrier (release h_t, acquire everyone else's) ----
    __threadfence();          // release: make h_t stores device-visible
    __syncthreads();
    if (threadIdx.x == 0) {
      unsigned prev = atomicAdd(arrive, 1u);
      if (prev == nblk - 1u) {
        atomicExch(arrive, 0u);
        __threadfence();
        atomicAdd(gen, 1u);
      } else {
        while (atomicAdd(gen, 0u) < (unsigned)(t + 1)) {
          __builtin_amdgcn_s_sleep(2);
        }
      }
    }
    __syncthreads();
    __threadfence();          // acquire: invalidate caches before reading h_t
  }
}

// ---------------------------------------------------------------------------
extern "C" void kernel_launch(void* const* d_in, const int* in_sizes, int n_in,
                              void* d_out, int out_size, void* d_ws,
                              size_t ws_size, hipStream_t stream) {
  const float* x  = (const float*)d_in[0];  // (B,S,IN)
  const float* h0 = (const float*)d_in[1];  // (B,OUT) zeros
  const float* W  = (const float*)d_in[2];  // (OUT,IN)
  const float* b  = (const float*)d_in[3];  // (OUT,)
  const float* U  = (const float*)d_in[4];  // (OUT,OUT)
  float* out      = (float*)d_out;          // (B,S,OUT)

  // Zero the 2-word grid-barrier state each call (capturable memset node).
  hipMemsetAsync(d_ws, 0, 2 * sizeof(unsigned), stream);

  // Phase 1: x2h into d_out.  M = B*S rows, 64 rows per block-row.
  x2h_kernel<<<dim3(8, (BDIM * SDIM) / 64), dim3(256), 0, stream>>>(x, W, b,
                                                                    out);
  // Phase 2: persistent sequential scan over S.
  rnn_scan_kernel<<<dim3(NBLK_SCAN), dim3(256), 0, stream>>>(
      h0, U, out, (unsigned*)d_ws);
}